// NHP_25537875542458
// MI455X (gfx1250) — compile-verified
//
#include <hip/hip_runtime.h>
#include <hip/hip_bf16.h>
#include <cfloat>

typedef __attribute__((ext_vector_type(2))) float v2f;
typedef __attribute__((ext_vector_type(8))) float v8f;

#define IN_CH 128
#define INC_PER_EDGE 12

// ---------------------------------------------------------------------------
// Kernel 0: pack B matrices into WMMA fragment order + fuse biases.
//   Bpack index = ((t*32 + s)*32 + lane)*2 + v  holds  Wt[k][n] = W[n][k]
//   with n = 16t + (lane&15), k = 4s + 2*(lane>>4) + v.
//   Bd = (W_self - W_hyp) packed, Bh = W_hyp packed, bias = b_self + b_hyp.
// ---------------------------------------------------------------------------
__global__ void nhp_prep_kernel(const float* __restrict__ W_self,
                                const float* __restrict__ b_self,
                                const float* __restrict__ W_hyp,
                                const float* __restrict__ b_hyp,
                                float* __restrict__ Bd,
                                float* __restrict__ Bh,
                                float* __restrict__ bias) {
  for (int i = threadIdx.x; i < IN_CH * IN_CH; i += blockDim.x) {
    int v    = i & 1;
    int lane = (i >> 1) & 31;
    int s    = (i >> 6) & 31;
    int t    = i >> 11;
    int kh   = lane >> 4;
    int n    = 16 * t + (lane & 15);
    int k    = 4 * s + 2 * kh + v;
    float wh = W_hyp[n * IN_CH + k];
    float ws = W_self[n * IN_CH + k];
    Bd[i] = ws - wh;
    Bh[i] = wh;
  }
  if (threadIdx.x < IN_CH)
    bias[threadIdx.x] = b_self[threadIdx.x] + b_hyp[threadIdx.x];
}

// ---------------------------------------------------------------------------
// Kernel 1: he_sum[e][c] = sum_{j<12} x[node_ids[12e+j]][c]
// Incidences of edge e are contiguous (he_ids = arange // 12): no atomics.
// 128 threads = channels -> coalesced 512B row reads.
// ---------------------------------------------------------------------------
__global__ void nhp_hesum_kernel(const float* __restrict__ x,
                                 const int* __restrict__ node_ids,
                                 float* __restrict__ he_sum, int E) {
  int e = blockIdx.x;
  if (e >= E) return;
  int c = threadIdx.x;
  const int* ids = node_ids + e * INC_PER_EDGE;
  float s = 0.0f;
#pragma unroll
  for (int j = 0; j < INC_PER_EDGE; ++j) {
    int n = ids[j];  // wave-uniform -> scalar load
    s += x[(size_t)n * IN_CH + c];
  }
  he_sum[(size_t)e * IN_CH + c] = s;
}

// ---------------------------------------------------------------------------
// Kernel 2: Y = X @ W^T via V_WMMA_F32_16X16X4_F32 (full fp32 precision).
// One wave computes one 16-row x 128-col tile: 8 accumulators, 32 K-steps.
// B fragments read from pre-packed, wave-coalesced layout (stays in L0/L2).
// Early exit is wave-uniform => EXEC all-1s at every WMMA (ISA requirement).
// ---------------------------------------------------------------------------
__global__ void __launch_bounds__(128)
nhp_gemm_kernel(const float* __restrict__ X, const float* __restrict__ Bpack,
                float* __restrict__ Y, int R) {
  int lane = threadIdx.x & 31;
  int wave = threadIdx.x >> 5;
  int gw   = blockIdx.x * (blockDim.x >> 5) + wave;
  int r0   = gw << 4;
  if (r0 >= R) return;                 // wave-uniform guard

  int m  = lane & 15;
  int kh = lane >> 4;                  // 0 or 1

  const float* xrow  = X + (size_t)(r0 + m) * IN_CH + 2 * kh;
  const float* bbase = Bpack + lane * 2;

  v8f acc[8];
#pragma unroll
  for (int t = 0; t < 8; ++t)
    acc[t] = (v8f){0.f, 0.f, 0.f, 0.f, 0.f, 0.f, 0.f, 0.f};

  for (int s = 0; s < 32; ++s) {
    v2f a = *(const v2f*)(xrow + 4 * s);   // A frag: X[r0+m][4s+2kh .. +1]
#pragma unroll
    for (int t = 0; t < 8; ++t) {          // 8 independent WMMA chains
      v2f b = *(const v2f*)(bbase + (t * 32 + s) * 64);
      acc[t] = __builtin_amdgcn_wmma_f32_16x16x4_f32(
          /*neg_a=*/false, a, /*neg_b=*/false, b,
          /*c_mod=*/(short)0, acc[t], /*reuse_a=*/false, /*reuse_b=*/false);
    }
  }

  // D layout: acc[t][v] = Y[r0 + v + 8*kh][16t + (lane&15)]
#pragma unroll
  for (int t = 0; t < 8; ++t) {
#pragma unroll
    for (int v = 0; v < 8; ++v) {
      Y[(size_t)(r0 + v + 8 * kh) * IN_CH + 16 * t + m] = acc[t][v];
    }
  }
}

// ---------------------------------------------------------------------------
// Kernel 3: per edge e (one wave, 4 channels/lane, float4 gathers):
//   emb_j = relu(y_node[nid_j] + y_he[e] + bias); he_emb = max_j - min_j;
//   score = dot(he_emb, W_score) + b_score  (wave shfl_xor reduction).
// ---------------------------------------------------------------------------
__global__ void __launch_bounds__(256)
nhp_score_kernel(const float* __restrict__ y_node,
                 const float* __restrict__ y_he,
                 const int* __restrict__ node_ids,
                 const float* __restrict__ bias,
                 const float* __restrict__ W_score,
                 const float* __restrict__ b_score,
                 float* __restrict__ out, int E) {
  int lane = threadIdx.x & 31;
  int wave = threadIdx.x >> 5;
  int e = blockIdx.x * (blockDim.x >> 5) + wave;
  if (e >= E) return;

  int c0 = lane * 4;
  float4 yh = *(const float4*)(y_he + (size_t)e * IN_CH + c0);
  float4 bb = *(const float4*)(bias + c0);
  float base_x = yh.x + bb.x, base_y = yh.y + bb.y;
  float base_z = yh.z + bb.z, base_w = yh.w + bb.w;

  float mx_x = -FLT_MAX, mx_y = -FLT_MAX, mx_z = -FLT_MAX, mx_w = -FLT_MAX;
  float mn_x =  FLT_MAX, mn_y =  FLT_MAX, mn_z =  FLT_MAX, mn_w =  FLT_MAX;

  const int* ids = node_ids + e * INC_PER_EDGE;
#pragma unroll
  for (int j = 0; j < INC_PER_EDGE; ++j) {
    int n = ids[j];  // wave-uniform
    float4 yn = *(const float4*)(y_node + (size_t)n * IN_CH + c0);
    float tx = fmaxf(yn.x + base_x, 0.0f);
    float ty = fmaxf(yn.y + base_y, 0.0f);
    float tz = fmaxf(yn.z + base_z, 0.0f);
    float tw = fmaxf(yn.w + base_w, 0.0f);
    mx_x = fmaxf(mx_x, tx); mn_x = fminf(mn_x, tx);
    mx_y = fmaxf(mx_y, ty); mn_y = fminf(mn_y, ty);
    mx_z = fmaxf(mx_z, tz); mn_z = fminf(mn_z, tz);
    mx_w = fmaxf(mx_w, tw); mn_w = fminf(mn_w, tw);
  }

  float4 ws = *(const float4*)(W_score + c0);
  float p = (mx_x - mn_x) * ws.x + (mx_y - mn_y) * ws.y +
            (mx_z - mn_z) * ws.z + (mx_w - mn_w) * ws.w;
#pragma unroll
  for (int off = 16; off > 0; off >>= 1)
    p += __shfl_xor(p, off, 32);
  if (lane == 0) out[e] = p + b_score[0];
}

// ---------------------------------------------------------------------------
extern "C" void kernel_launch(void* const* d_in, const int* in_sizes, int n_in,
                              void* d_out, int out_size, void* d_ws, size_t ws_size,
                              hipStream_t stream) {
  const float* x        = (const float*)d_in[0];
  const int*   hidx     = (const int*)d_in[1];   // [2, num_inc]; row0 = node_ids
  const float* W_self   = (const float*)d_in[2];
  const float* b_self   = (const float*)d_in[3];
  const float* W_hyp    = (const float*)d_in[4];
  const float* b_hyp    = (const float*)d_in[5];
  const float* W_score  = (const float*)d_in[6];
  const float* b_score  = (const float*)d_in[7];
  float* out = (float*)d_out;

  const int num_inc = in_sizes[1] / 2;
  const int E = num_inc / INC_PER_EDGE;          // 50000
  const int N = in_sizes[0] / IN_CH;             // 100000
  const int* node_ids = hidx;                    // first num_inc ints

  // workspace layout (floats)
  float* ws      = (float*)d_ws;
  size_t off = 0;
  float* he_sum  = ws + off; off += (size_t)E * IN_CH;   // 25.6 MB
  float* y_node  = ws + off; off += (size_t)N * IN_CH;   // 51.2 MB
  float* y_he    = ws + off; off += (size_t)E * IN_CH;   // 25.6 MB
  float* Bd      = ws + off; off += IN_CH * IN_CH;       // 64 KB
  float* Bh      = ws + off; off += IN_CH * IN_CH;       // 64 KB
  float* bias    = ws + off; off += IN_CH;

  nhp_prep_kernel<<<1, 256, 0, stream>>>(W_self, b_self, W_hyp, b_hyp, Bd, Bh, bias);

  nhp_hesum_kernel<<<E, IN_CH, 0, stream>>>(x, node_ids, he_sum, E);

  // y_node = x @ (W_self - W_hyp)^T   : N x 128
  {
    int grid = (N + 63) / 64;   // 64 rows per block (4 waves x 16 rows)
    nhp_gemm_kernel<<<grid, 128, 0, stream>>>(x, Bd, y_node, N);
  }
  // y_he = he_sum @ W_hyp^T           : E x 128
  {
    int grid = (E + 63) / 64;
    nhp_gemm_kernel<<<grid, 128, 0, stream>>>(he_sum, Bh, y_he, E);
  }

  // fused gather + relu + max-min pool + score
  {
    int grid = (E + 7) / 8;     // 8 edges per 256-thread block
    nhp_score_kernel<<<grid, 256, 0, stream>>>(y_node, y_he, node_ids, bias,
                                               W_score, b_score, out, E);
  }
}